// CapsNet_29970281791626
// MI455X (gfx1250) — compile-verified
//
#include <hip/hip_runtime.h>
#include <math.h>

// ---------------------------------------------------------------------------
// CapsNet EM-routing conv-caps forward for MI455X (gfx1250, wave32).
//   b=4, h=w=10, K=3 -> oh=ow=8, bb=256, BB=288, C=32, PSIZE=16 (P=4).
// Pipeline:
//   1) gather  : reproduce reference's (scrambled) patch reshape into
//                p_in (256,288,16) and a_in (256,288) in d_ws.
//   2) votes   : v[bb,n,c,ik] = sum_j x4[bb,n,i,j] * W[n,c,j,k] via
//                V_WMMA_F32_16X16X4_F32 (K=4 == P, f32-exact, dense tiles).
//                v is stored in WMMA-tile-native layout
//                   v_t[bbgrp][n][ct][q][lane]   (bbgrp=bb/4, ct=c/4)
//                so every D-register store is a fully coalesced 128B/wave
//                global_store_b32 (vs 16B chunks for a (c,ps) layout).
//   3) routing : 3 EM iterations per bb; r/coeff, mu, sigma in ~42KB LDS.
//                mu and sigma fused into ONE pass over v using
//                   sigma = sum(c*v^2) - mu^2*(2 - s) + eps,  s = rs/(rs+eps)
//                (algebraically identical to the reference), cutting v
//                traffic from 8 to 5 full passes (~0.76 GB, L2-resident:
//                v = 151 MB < 192 MB L2).
// Workspace requirement: ~156 MB.
// ---------------------------------------------------------------------------

#define NBB   256      // b*oh*ow
#define NN    288      // K*K*B_CAPS
#define NC    32       // C_CAPS
#define NPS   16       // PSIZE
#define EPSF  1e-8f
#define LAMF  1e-6f
#define LN2PI 1.8378770664093453f

// v tile-native layout constants: per (bbgrp,n) tile region = 8 ct * 256
#define VTILE 2048     // floats per (bbgrp, n)

typedef __attribute__((ext_vector_type(2))) float v2f;
typedef __attribute__((ext_vector_type(4))) float v4f;
typedef __attribute__((ext_vector_type(8))) float v8f;

// Offset of element (c, ps) for sub-position bbl inside a (bbgrp,n) tile.
// D-tile layout: VGPR q holds M = q + 8*(lane/16), N = lane%16,
//   M = 4*bbl + i, N = 4*cl + k, stored at ct*256 + q*32 + lane.
__device__ __forceinline__ int vt_off(int bbl, int c, int ps) {
    const int i  = ps >> 2, k = ps & 3;
    const int ct = c >> 2, cl = c & 3;
    const int m  = bbl * 4 + i;
    return ct * 256 + (m & 7) * 32 + (m >> 3) * 16 + cl * 4 + k;
}

// ---------------------------------------------------------------------------
// Kernel 1: patch gather, faithful to the reference's transpose+reshape.
// p_in flat index f = 4608*bb + 16*n + ps maps to (b,ki,kj,i,j,ch) coords of
// the (4,3,3,8,8,512) tensor; a_in flat index g = 288*bb + n maps into
// (4,3,3,8,8,32).  x is (4,10,10,544): first 512 ch = pose, last 32 = act.
// ---------------------------------------------------------------------------
__global__ void caps_gather_kernel(const float* __restrict__ x,
                                   float* __restrict__ p_in,
                                   float* __restrict__ a_in) {
    int t = blockIdx.x * 256 + threadIdx.x;
    const int NP = NBB * NN * NPS;          // 1179648
    if (t < NP) {
        int f  = t;
        int ch = f & 511;
        int j  = (f >> 9) & 7;
        int i  = (f >> 12) & 7;
        int kj = (f / 32768) % 3;
        int ki = (f / 98304) % 3;
        int b  =  f / 294912;
        p_in[t] = x[((b * 10 + ki + i) * 10 + kj + j) * 544 + ch];
    } else if (t < NP + NBB * NN) {
        int g   = t - NP;
        int c32 = g & 31;
        int j   = (g >> 5) & 7;
        int i   = (g >> 8) & 7;
        int kj  = (g / 2048) % 3;
        int ki  = (g / 6144) % 3;
        int b   =  g / 18432;
        a_in[g] = x[((b * 10 + ki + i) * 10 + kj + j) * 544 + 512 + c32];
    }
}

// ---------------------------------------------------------------------------
// Kernel 2: votes via WMMA f32 16x16x4.
//   Per n (independent 4x4 weight blocks): GEMM  A(16x4) * B(4x16) -> D(16x16)
//     A rows m = 4*bb_local + i ; cols = j (K=4)
//     B rows j ; cols nn = 4*c_local + k
//   VGPR layout (ISA 7.12.2, 32-bit A 16x4): lanes 0-15 hold K=0 (v0), K=1
//   (v1); lanes 16-31 hold K=2 (v0), K=3 (v1).  B mirrors for its K rows.
// Grid sized to exactly 64*288 waves -> EXEC all ones (WMMA requirement).
// D stored tile-native: v_t[(bbgrp*NN+n)*VTILE + ct*256 + q*32 + lane]
//   -> 8 perfectly coalesced b32 stores per WMMA wave-wide.
// ---------------------------------------------------------------------------
__global__ void caps_votes_kernel(const float* __restrict__ p_in,
                                  const float* __restrict__ W,   // (288,32,4,4)
                                  float* __restrict__ v) {       // tile-native
    const int wave = (blockIdx.x * blockDim.x + threadIdx.x) >> 5;
    const int lane = threadIdx.x & 31;
    const int n    = wave % NN;
    const int grp  = wave / NN;                // bb group of 4
    const int half = lane >> 4;                // 0: K=0,1   1: K=2,3
    const int l    = lane & 15;

    // A operand: row m = l, j = 2*half + {0,1}  (8B aligned -> float2 load)
    const int bbA = grp * 4 + (l >> 2);
    const int iA  = l & 3;
    const v2f A = *(const v2f*)(p_in + ((bbA * NN + n) * NPS) + iA * 4 + half * 2);

    const int k  = l & 3;        // output pose col
    const int cl = l >> 2;       // c within tile
    const int j0 = half * 2;

    float* vt = v + ((size_t)grp * NN + n) * VTILE + lane;

    for (int ct = 0; ct < 8; ++ct) {
        const int c = ct * 4 + cl;
        // B operand: B[j, nn] = W[n, c, j, k]
        const float* bp = W + ((n * NC + c) * 4) * 4 + k;
        v2f Bv;
        Bv.x = bp[(j0 + 0) * 4];
        Bv.y = bp[(j0 + 1) * 4];

        v8f D = {};
        D = __builtin_amdgcn_wmma_f32_16x16x4_f32(
                /*neg_a=*/false, A, /*neg_b=*/false, Bv,
                /*c_mod=*/(short)0, D, /*reuse_a=*/false, /*reuse_b=*/false);

        #pragma unroll
        for (int q = 0; q < 8; ++q)
            vt[ct * 256 + q * 32] = D[q];
    }
}

// ---------------------------------------------------------------------------
// Kernel 3: EM routing, one block (256 threads) per bb.
// ---------------------------------------------------------------------------
__global__ void __launch_bounds__(256)
caps_routing_kernel(const float* __restrict__ v_g,   // tile-native
                    const float* __restrict__ a_in_g,
                    const float* __restrict__ beta_u,   // (32,16)
                    const float* __restrict__ beta_a,   // (32)
                    float* __restrict__ out) {          // (256,544)
    __shared__ float r_s[NN * NC];       // r / coeff / ln_ap (reused)
    __shared__ float ain_s[NN];
    __shared__ float mu_s[NC * NPS];
    __shared__ float sig_s[NC * NPS];
    __shared__ float rsum_s[NC];
    __shared__ float cost_s[NC];
    __shared__ float aout_s[NC];
    __shared__ float scal_s[2];          // {cost_mean, cost_stdv}

    const int bb  = blockIdx.x;
    const int bbl = bb & 3;
    const int tid = threadIdx.x;
    const float* vbase = v_g + (size_t)(bb >> 2) * NN * VTILE;

    for (int n = tid; n < NN; n += 256) ain_s[n] = a_in_g[bb * NN + n];
    for (int t = tid; t < NN * NC; t += 256) r_s[t] = 1.0f / 32.0f;
    __syncthreads();

    // Per-thread (c,ps) slots for the mu/sigma pass.
    const int i0 = tid, i1 = tid + 256;
    const int c0 = i0 >> 4, c1 = i1 >> 4;
    const int o0 = vt_off(bbl, c0, i0 & 15);
    const int o1 = vt_off(bbl, c1, i1 & 15);

    for (int it = 0; it < 3; ++it) {
        // ---- m-step ----
        // r *= a_in ; normalize over c (per n)
        for (int n = tid; n < NN; n += 256) {
            float* rr = r_s + n * NC;
            const float ai = ain_s[n];
            float s = 0.0f;
            #pragma unroll
            for (int c = 0; c < NC; ++c) { float t2 = rr[c] * ai; rr[c] = t2; s += t2; }
            const float inv = 1.0f / (s + EPSF);
            #pragma unroll
            for (int c = 0; c < NC; ++c) rr[c] *= inv;
        }
        __syncthreads();
        // r_sum[c] = sum_n r[n,c]
        if (tid < NC) {
            float s = 0.0f;
            for (int n = 0; n < NN; ++n) s += r_s[n * NC + tid];
            rsum_s[tid] = s;
        }
        __syncthreads();
        // coeff = r / (r_sum + eps)   (in place)
        for (int t = tid; t < NN * NC; t += 256)
            r_s[t] = r_s[t] / (rsum_s[t & 31] + EPSF);
        __syncthreads();
        // Fused mu/sigma single pass over v:
        //   mu  = sum coeff*v
        //   sig = sum coeff*v^2 - mu^2*(2 - s) + eps,  s = sum coeff
        {
            float a0 = 0.0f, b0 = 0.0f, a1 = 0.0f, b1 = 0.0f;
            for (int n = 0; n < NN; ++n) {
                const float* vn = vbase + (size_t)n * VTILE;
                const float x0 = vn[o0], x1 = vn[o1];
                const float w0 = r_s[n * NC + c0], w1 = r_s[n * NC + c1];
                a0 += w0 * x0; b0 += w0 * x0 * x0;
                a1 += w1 * x1; b1 += w1 * x1 * x1;
            }
            const float s0 = rsum_s[c0] / (rsum_s[c0] + EPSF);
            const float s1 = rsum_s[c1] / (rsum_s[c1] + EPSF);
            mu_s[i0]  = a0;                       mu_s[i1]  = a1;
            sig_s[i0] = b0 - a0 * a0 * (2.0f - s0) + EPSF;
            sig_s[i1] = b1 - a1 * a1 * (2.0f - s1) + EPSF;
        }
        __syncthreads();
        // cost_h[c] = sum_ps (beta_u + 0.5*log sigma) * r_sum[c]
        if (tid < NC) {
            float s = 0.0f;
            #pragma unroll
            for (int ps = 0; ps < NPS; ++ps)
                s += beta_u[tid * NPS + ps] + 0.5f * __logf(sig_s[tid * NPS + ps]);
            cost_s[tid] = s * rsum_s[tid];
        }
        __syncthreads();
        // cost_mean, cost_stdv (reference quirk: sum THEN square)
        if (tid == 0) {
            float s = 0.0f;
            for (int c = 0; c < NC; ++c) s += cost_s[c];
            const float mean = s * (1.0f / NC);
            float d = 0.0f;
            for (int c = 0; c < NC; ++c) d += cost_s[c] - mean;
            scal_s[0] = mean;
            scal_s[1] = sqrtf(d * d * (1.0f / NC) + EPSF);
        }
        __syncthreads();
        // a_out[c]
        if (tid < NC) {
            const float mean = scal_s[0], stdv = scal_s[1];
            const float z = LAMF * (beta_a[tid] - (mean - cost_s[tid]) / (stdv + EPSF));
            aout_s[tid] = 1.0f / (1.0f + __expf(-z));
        }
        __syncthreads();

        // ---- e-step (skipped on last iteration) ----
        if (it < 2) {
            for (int t = tid; t < NN * NC; t += 256) {
                const int n = t >> 5, c = t & 31;
                const float* vb = vbase + (size_t)n * VTILE + (c >> 2) * 256 + (c & 3) * 4;
                float s = 0.0f;
                #pragma unroll
                for (int i = 0; i < 4; ++i) {
                    const int m = bbl * 4 + i;
                    const v4f vv = *(const v4f*)(vb + (m & 7) * 32 + (m >> 3) * 16);
                    #pragma unroll
                    for (int k = 0; k < 4; ++k) {
                        const int ps = i * 4 + k;
                        const float sg = sig_s[c * NPS + ps];
                        const float d  = vv[k] - mu_s[c * NPS + ps];
                        s += -d * d / (2.0f * sg) - 0.5f * __logf(sg) - 0.5f * LN2PI;
                    }
                }
                r_s[t] = s + __logf(EPSF + aout_s[c]);   // coeff no longer needed
            }
            __syncthreads();
            // softmax over c per n
            for (int n = tid; n < NN; n += 256) {
                float* rr = r_s + n * NC;
                float mx = rr[0];
                #pragma unroll
                for (int c = 1; c < NC; ++c) mx = fmaxf(mx, rr[c]);
                float s = 0.0f;
                #pragma unroll
                for (int c = 0; c < NC; ++c) { float e = __expf(rr[c] - mx); rr[c] = e; s += e; }
                const float inv = 1.0f / s;
                #pragma unroll
                for (int c = 0; c < NC; ++c) rr[c] *= inv;
            }
            __syncthreads();
        }
    }

    // output: (b,8,8, 512 pose | 32 act), row stride 544
    float* o = out + (size_t)bb * 544;
    o[tid]       = mu_s[tid];
    o[tid + 256] = mu_s[tid + 256];
    if (tid < NC) o[512 + tid] = aout_s[tid];
}

// ---------------------------------------------------------------------------
extern "C" void kernel_launch(void* const* d_in, const int* in_sizes, int n_in,
                              void* d_out, int out_size, void* d_ws, size_t ws_size,
                              hipStream_t stream) {
    (void)in_sizes; (void)n_in; (void)out_size; (void)ws_size;

    const float* x      = (const float*)d_in[0];   // (4,10,10,544)
    const float* beta_u = (const float*)d_in[1];   // (32,16)
    const float* beta_a = (const float*)d_in[2];   // (32,)
    const float* wts    = (const float*)d_in[3];   // (1,288,32,4,4)
    float* out          = (float*)d_out;

    char* ws = (char*)d_ws;
    float* p_in = (float*)ws;                                   //  4.72 MB
    float* a_in = (float*)(ws + (size_t)NBB * NN * NPS * 4);    //  0.30 MB
    float* v    = (float*)(ws + (size_t)NBB * NN * NPS * 4
                              + (size_t)NBB * NN * 4);          // 151.0 MB

    // 1) gather
    {
        const int total = NBB * NN * NPS + NBB * NN;            // 1,253,376
        caps_gather_kernel<<<(total + 255) / 256, 256, 0, stream>>>(x, p_in, a_in);
    }
    // 2) votes (WMMA): 64 bb-groups * 288 n = 18432 waves, 8 waves/block
    {
        const int blocks = (64 * NN) / 8;                       // 2304
        caps_votes_kernel<<<blocks, 256, 0, stream>>>(p_in, wts, v);
    }
    // 3) routing
    caps_routing_kernel<<<NBB, 256, 0, stream>>>(v, a_in, beta_u, beta_a, out);
}